// LogAttention1d_67594195304982
// MI455X (gfx1250) — compile-verified
//
#include <hip/hip_runtime.h>
#include <math.h>

typedef __attribute__((ext_vector_type(2))) float v2f;
typedef __attribute__((ext_vector_type(8))) float v8f;

#define B_   4
#define C_   256
#define WLEN 3125
#define LP   4096   // padded stride for the residual accumulator (= Lmax)
#define NH_  8
#define KD_  64
#define HD_  32

// ---------------- BN fold: scale = g*rsqrt(v+eps), bias = b - m*scale ----------------
__global__ void bn_prep_kernel(const float* __restrict__ g, const float* __restrict__ b,
                               const float* __restrict__ m, const float* __restrict__ v,
                               int n, float* __restrict__ sc, float* __restrict__ bi) {
    int i = blockIdx.x * blockDim.x + threadIdx.x;
    if (i < n) {
        float s = g[i] * rsqrtf(v[i] + 1e-5f);
        sc[i] = s;
        bi[i] = b[i] - m[i] * s;
    }
}

// Copy x (B,C,WLEN) into padded xa (B,C,LP), zero beyond WLEN.
__global__ void copy_pad_kernel(const float* __restrict__ src, float* __restrict__ dst) {
    int i = blockIdx.x * blockDim.x + threadIdx.x;
    if (i >= B_ * C_ * LP) return;
    int l  = i % LP;
    int bc = i / LP;
    dst[i] = (l < WLEN) ? src[(size_t)bc * WLEN + l] : 0.0f;
}

// ---------------- fp32 WMMA GEMM + fused BN, software-pipelined ----------------
// Y[bb, o, l] = BN( sum_c Wm[o,c] * X[bb, c, l] )
// X pre-padded (stride ldx, zeros beyond valid) -> branch-free inner loop.
// Wave: 16 (M) x 64 (N: 4 WMMA tiles reusing one A fragment). Block: 8 waves -> 128x64.

struct Frag {
    v2f a0, a1;      // A fragments for two K-steps
    v2f b0[4], b1[4];// B fragments for 4 N-tiles, two K-steps
};

__device__ __forceinline__ void load_frag(Frag& f, const float* __restrict__ Wr,
                                          const float* __restrict__ Xb, int ldx,
                                          int kk, int khi) {
    f.a0 = *(const v2f*)(Wr + kk);
    f.a1 = *(const v2f*)(Wr + kk + 4);
    const float* p0 = Xb + (size_t)(kk + 2 * khi) * ldx;
    const float* p1 = p0 + (size_t)4 * ldx;
#pragma unroll
    for (int t = 0; t < 4; ++t) {
        f.b0[t].x = p0[t * 16];
        f.b0[t].y = p0[(size_t)ldx + t * 16];
        f.b1[t].x = p1[t * 16];
        f.b1[t].y = p1[(size_t)ldx + t * 16];
    }
}

__device__ __forceinline__ void mma_frag(const Frag& f, v8f acc[4]) {
#pragma unroll
    for (int t = 0; t < 4; ++t)
        acc[t] = __builtin_amdgcn_wmma_f32_16x16x4_f32(
            false, f.a0, false, f.b0[t], (short)0, acc[t], false, false);
#pragma unroll
    for (int t = 0; t < 4; ++t)
        acc[t] = __builtin_amdgcn_wmma_f32_16x16x4_f32(
            false, f.a1, false, f.b1[t], (short)0, acc[t], false, false);
}

__global__ __launch_bounds__(256)
void gemm_bn_wmma(const float* __restrict__ X, int ldx,
                  const float* __restrict__ Wm, int O,
                  const float* __restrict__ sc, const float* __restrict__ bi,
                  float* __restrict__ Y, int ldy, int Lout) {
    const int lane = threadIdx.x & 31;
    const int wave = threadIdx.x >> 5;
    const int bb   = blockIdx.z;
    const int o_base = blockIdx.y * 128 + wave * 16;
    const int l_base = blockIdx.x * 64;
    const int mrow = lane & 15;   // M row within tile (and N col for B/D)
    const int khi  = lane >> 4;   // K-half select per WMMA f32 16x16x4 layout

    v8f acc[4] = {};
    const float* Xb = X + (size_t)bb * C_ * (size_t)ldx + l_base + mrow;
    const float* Wr = Wm + (size_t)(o_base + mrow) * C_ + 2 * khi;

    Frag f0, f1;
    load_frag(f0, Wr, Xb, ldx, 0, khi);
    // Pipelined K loop: 16 K per iteration, one fragment-group lookahead.
    for (int kk = 0; kk < C_; kk += 16) {
        load_frag(f1, Wr, Xb, ldx, kk + 8, khi);
        mma_frag(f0, acc);
        if (kk + 16 < C_) load_frag(f0, Wr, Xb, ldx, kk + 16, khi);
        mma_frag(f1, acc);
    }

#pragma unroll
    for (int t = 0; t < 4; ++t) {
        const int ln = l_base + t * 16 + mrow;
        if (ln >= Lout) continue;
#pragma unroll
        for (int r = 0; r < 8; ++r) {
            const int o = o_base + r + 8 * khi;  // D layout: VGPR r -> M = r + 8*(lane/16)
            Y[((size_t)bb * O + o) * (size_t)ldy + ln] = acc[t][r] * sc[o] + bi[o];
        }
    }
}

// ---------------- one log-attention level (base x base softmax) ----------------
// qk layout: (b, NH, {q,k}, KD, L) flattened -> channel = h*128 + t*64 + d
// v   layout: (b, NH, HD, L)   -> channel = h*32 + d
// group g over (i0, k): l(j) = i0*BASE*s2 + j*s2 + k ; softmax over j per output row J.
template <int BASE>
__global__ __launch_bounds__(256)
void attn_level_kernel(const float* __restrict__ qk,
                       const float* __restrict__ vin,
                       float* __restrict__ vout,
                       int L, int s2, int G) {
    const int tid = blockIdx.x * blockDim.x + threadIdx.x;
    const int total = B_ * NH_ * G;
    if (tid >= total) return;
    const int g  = tid % G;
    const int bh = tid / G;
    const int i0 = g / s2;
    const int k  = g - i0 * s2;
    const int l0 = i0 * BASE * s2 + k;

    const float* qp = qk + (size_t)bh * 128 * (size_t)L + l0;          // q: t=0
    const float* kp = qp + (size_t)KD_ * (size_t)L;                    // k: t=1

    float S[BASE][BASE];  // [J][j]
#pragma unroll
    for (int J = 0; J < BASE; ++J)
#pragma unroll
        for (int j = 0; j < BASE; ++j) S[J][j] = 0.0f;

    for (int d = 0; d < KD_; ++d) {
        float qv[BASE], kv[BASE];
#pragma unroll
        for (int j = 0; j < BASE; ++j) qv[j] = qp[(size_t)d * L + (size_t)j * s2];
#pragma unroll
        for (int J = 0; J < BASE; ++J) kv[J] = kp[(size_t)d * L + (size_t)J * s2];
#pragma unroll
        for (int J = 0; J < BASE; ++J)
#pragma unroll
            for (int j = 0; j < BASE; ++j) S[J][j] = fmaf(kv[J], qv[j], S[J][j]);
    }

    const float scale = 0.125f;  // KD^-0.5 = 64^-0.5
#pragma unroll
    for (int J = 0; J < BASE; ++J) {
        float mx = S[J][0];
#pragma unroll
        for (int j = 1; j < BASE; ++j) mx = fmaxf(mx, S[J][j]);
        float sum = 0.0f;
#pragma unroll
        for (int j = 0; j < BASE; ++j) {
            float e = __expf((S[J][j] - mx) * scale);
            S[J][j] = e;
            sum += e;
        }
        const float inv = 1.0f / sum;
#pragma unroll
        for (int j = 0; j < BASE; ++j) S[J][j] *= inv;
    }

    const float* vp = vin  + (size_t)bh * HD_ * (size_t)L + l0;
    float*       op = vout + (size_t)bh * HD_ * (size_t)L + l0;
    for (int d = 0; d < HD_; ++d) {
        float vv[BASE];
#pragma unroll
        for (int j = 0; j < BASE; ++j) vv[j] = vp[(size_t)d * L + (size_t)j * s2];
#pragma unroll
        for (int J = 0; J < BASE; ++J) {
            float o = 0.0f;
#pragma unroll
            for (int j = 0; j < BASE; ++j) o = fmaf(S[J][j], vv[j], o);
            op[(size_t)d * L + (size_t)J * s2] = o;
        }
    }
}

// ---------------- depthwise-3 conv + BN + residual accumulate (into padded xa) ----------------
__global__ __launch_bounds__(256)
void pe_residual_kernel(const float* __restrict__ v2, const float* __restrict__ v0,
                        const float* __restrict__ peW,  // [256][3]
                        const float* __restrict__ sc, const float* __restrict__ bi,
                        float* __restrict__ xa, int L) {
    const int idx = blockIdx.x * blockDim.x + threadIdx.x;
    if (idx >= B_ * C_ * WLEN) return;
    const int l  = idx % WLEN;
    const int bc = idx / WLEN;
    const int ch = bc % C_;
    const size_t base = (size_t)bc * L + l;
    const float* w3 = peW + ch * 3;
    float acc = v0[base] * w3[1];
    if (l > 0)     acc = fmaf(v0[base - 1], w3[0], acc);
    if (l + 1 < L) acc = fmaf(v0[base + 1], w3[2], acc);
    const float pe = acc * sc[ch] + bi[ch];
    xa[(size_t)bc * LP + l] += v2[base] + pe;
}

extern "C" void kernel_launch(void* const* d_in, const int* in_sizes, int n_in,
                              void* d_out, int out_size, void* d_ws, size_t ws_size,
                              hipStream_t stream) {
    const float* x      = (const float*)d_in[0];
    const float* qk_W   = (const float*)d_in[1];
    const float* qk_g   = (const float*)d_in[2];
    const float* qk_b   = (const float*)d_in[3];
    const float* qk_m   = (const float*)d_in[4];
    const float* qk_v   = (const float*)d_in[5];
    const float* v_W    = (const float*)d_in[6];
    const float* v_g    = (const float*)d_in[7];
    const float* v_b    = (const float*)d_in[8];
    const float* v_m    = (const float*)d_in[9];
    const float* v_v    = (const float*)d_in[10];
    const float* pe_W   = (const float*)d_in[11];
    const float* pe_g   = (const float*)d_in[12];
    const float* pe_b   = (const float*)d_in[13];
    const float* pe_m   = (const float*)d_in[14];
    const float* pe_v   = (const float*)d_in[15];
    const float* proj_W = (const float*)d_in[16];
    const float* proj_g = (const float*)d_in[17];
    const float* proj_b = (const float*)d_in[18];
    const float* proj_m = (const float*)d_in[19];
    const float* proj_v = (const float*)d_in[20];

    float* ws    = (float*)d_ws;
    float* xa    = ws;                                  // B*C*LP (padded)
    float* qkb   = xa  + (size_t)B_ * C_ * LP;          // B*1024*LP
    float* v0    = qkb + (size_t)B_ * 1024 * LP;        // B*C*LP
    float* vA    = v0  + (size_t)B_ * C_ * LP;
    float* vB    = vA  + (size_t)B_ * C_ * LP;
    float* sc_qk = vB  + (size_t)B_ * C_ * LP;          // 1024
    float* bi_qk = sc_qk + 1024;
    float* sc_v  = bi_qk + 1024;                        // 256 each below
    float* bi_v  = sc_v + 256;
    float* sc_pe = bi_v + 256;
    float* bi_pe = sc_pe + 256;
    float* sc_p  = bi_pe + 256;
    float* bi_p  = sc_p + 256;

    copy_pad_kernel<<<(B_ * C_ * LP + 255) / 256, 256, 0, stream>>>(x, xa);

    const int bases[2] = {5, 4};
    const int nlev[2]  = {5, 6};
    const int Ls[2]    = {3125, 4096};
    const int ntot = B_ * C_ * WLEN;

    for (int idx = 0; idx < 2; ++idx) {
        const int base = bases[idx];
        const int n    = nlev[idx];
        const int L    = Ls[idx];

        bn_prep_kernel<<<4, 256, 0, stream>>>(qk_g + idx * 1024, qk_b + idx * 1024,
                                              qk_m + idx * 1024, qk_v + idx * 1024,
                                              1024, sc_qk, bi_qk);
        bn_prep_kernel<<<1, 256, 0, stream>>>(v_g + idx * 256, v_b + idx * 256,
                                              v_m + idx * 256, v_v + idx * 256,
                                              256, sc_v, bi_v);
        bn_prep_kernel<<<1, 256, 0, stream>>>(pe_g + idx * 256, pe_b + idx * 256,
                                              pe_m + idx * 256, pe_v + idx * 256,
                                              256, sc_pe, bi_pe);

        dim3 gq((L + 63) / 64, 1024 / 128, B_);
        gemm_bn_wmma<<<gq, 256, 0, stream>>>(xa, LP,
                                             qk_W + (size_t)idx * 1024 * 256, 1024,
                                             sc_qk, bi_qk, qkb, L, L);
        dim3 gv((L + 63) / 64, 256 / 128, B_);
        gemm_bn_wmma<<<gv, 256, 0, stream>>>(xa, LP,
                                             v_W + (size_t)idx * 256 * 256, 256,
                                             sc_v, bi_v, v0, L, L);

        const float* vin = v0;
        float* vout = vA;
        int s2 = L / base;          // base^(n-1)
        const int G = L / base;
        for (int i = 0; i < n; ++i) {
            const int blocks = (B_ * NH_ * G + 255) / 256;
            if (base == 5)
                attn_level_kernel<5><<<blocks, 256, 0, stream>>>(qkb, vin, vout, L, s2, G);
            else
                attn_level_kernel<4><<<blocks, 256, 0, stream>>>(qkb, vin, vout, L, s2, G);
            vin = vout;
            vout = (vout == vA) ? vB : vA;
            s2 /= base;
        }

        pe_residual_kernel<<<(ntot + 255) / 256, 256, 0, stream>>>(
            vin, v0, pe_W + (size_t)idx * C_ * 3, sc_pe, bi_pe, xa, L);
    }

    bn_prep_kernel<<<1, 256, 0, stream>>>(proj_g, proj_b, proj_m, proj_v, 256, sc_p, bi_p);
    dim3 gp((WLEN + 63) / 64, 256 / 128, B_);
    gemm_bn_wmma<<<gp, 256, 0, stream>>>(xa, LP, proj_W, 256,
                                         sc_p, bi_p, (float*)d_out, WLEN, WLEN);
}